// QLSTMCell_84980222919436
// MI455X (gfx1250) — compile-verified
//
#include <hip/hip_runtime.h>
#include <math.h>

typedef float v2f __attribute__((ext_vector_type(2)));
typedef float v8f __attribute__((ext_vector_type(8)));

#define BATCH  65536
#define INSZ   128
#define HID    512
#define COMB   640
#define WAVES  8

// workspace layout (float offsets)
#define WS_W    0          // stacked proj weights [16][640]
#define WS_PB   10240      // stacked proj bias [16]
#define WS_URE  10256      // U real  [4][16][16]  (row i = output state, col k)
#define WS_UIM  11280      // U imag  [4][16][16]
#define WS_MGT  12304      // M_g^T   [4][512][16] (row c = output col, col i = state)
#define WS_LB   45072      // stacked lin bias [4][512]

__device__ __forceinline__ v8f wmma4(v2f a, v2f b, v8f c) {
  return __builtin_amdgcn_wmma_f32_16x16x4_f32(false, a, false, b, (short)0, c, false, false);
}

// ---------------- setup: build U_g, stacked weights, M_g^T, biases ----------
__global__ void qlstm_setup(
    const float* pw0, const float* pb0, const float* qw0, const float* lw0, const float* lb0,
    const float* pw1, const float* pb1, const float* qw1, const float* lw1, const float* lb1,
    const float* pw2, const float* pb2, const float* qw2, const float* lw2, const float* lb2,
    const float* pw3, const float* pb3, const float* qw3, const float* lw3, const float* lb3,
    float* ws)
{
  const float* PW[4] = {pw0, pw1, pw2, pw3};
  const float* PB[4] = {pb0, pb1, pb2, pb3};
  const float* QW[4] = {qw0, qw1, qw2, qw3};
  const float* LW[4] = {lw0, lw1, lw2, lw3};
  const float* LB[4] = {lb0, lb1, lb2, lb3};
  const int tid = threadIdx.x;

  // stacked projection weights: Wstack[n][k] = pw_{n/4}[n%4][k]
  for (int i = tid; i < 16 * COMB; i += blockDim.x) {
    int n = i / COMB, k = i - n * COMB;
    ws[WS_W + i] = PW[n >> 2][(n & 3) * COMB + k];
  }
  if (tid < 16) ws[WS_PB + tid] = PB[tid >> 2][tid & 3];
  for (int i = tid; i < 4 * HID; i += blockDim.x)
    ws[WS_LB + i] = LB[i >> 9][i & (HID - 1)];

  // M_g^T[c][i] = sum_k zsign(i,k) * lin_w_g[c][k]
  for (int i = tid; i < 4 * HID * 16; i += blockDim.x) {
    int g = i >> 13;
    int r = i & 8191;
    int c = r >> 4;
    int ii = r & 15;
    const float* lw = LW[g] + c * 4;
    float acc = 0.f;
#pragma unroll
    for (int k = 0; k < 4; ++k) {
      float zs = ((ii >> (3 - k)) & 1) ? -1.f : 1.f;
      acc += zs * lw[k];
    }
    ws[WS_MGT + i] = acc;
  }

  // U_g: simulate the fixed part of the circuit on each basis column
  if (tid < 64) {
    int g = tid >> 4, col = tid & 15;
    const float* qw = QW[g];
    float sr[16], si[16];
    for (int i = 0; i < 16; ++i) { sr[i] = 0.f; si[i] = 0.f; }
    sr[col] = 1.f;
    for (int l = 0; l < 2; ++l) {
      // ring of CNOTs: new[i] = old[perm[i]]
      for (int j = 0; j < 4; ++j) {
        float tr[16], ti[16];
        int tq = (j + 1) & 3;
        for (int i = 0; i < 16; ++i) {
          int p = ((i >> (3 - j)) & 1) ? (i ^ (1 << (3 - tq))) : i;
          tr[i] = sr[p]; ti[i] = si[p];
        }
        for (int i = 0; i < 16; ++i) { sr[i] = tr[i]; si[i] = ti[i]; }
      }
      for (int j = 0; j < 4; ++j) {
        int mask = 1 << (3 - j);
        float tx = qw[(l * 4 + j) * 3 + 0];
        float ty = qw[(l * 4 + j) * 3 + 1];
        float tz = qw[(l * 4 + j) * 3 + 2];
        { // RX: [[c,-is],[-is,c]]
          float c = cosf(tx * 0.5f), s = sinf(tx * 0.5f);
          for (int i0 = 0; i0 < 16; ++i0) if (!(i0 & mask)) {
            int i1 = i0 | mask;
            float ar = sr[i0], ai = si[i0], br = sr[i1], bi = si[i1];
            sr[i0] = c * ar + s * bi; si[i0] = c * ai - s * br;
            sr[i1] = c * br + s * ai; si[i1] = c * bi - s * ar;
          }
        }
        { // RY: [[c,-s],[s,c]]
          float c = cosf(ty * 0.5f), s = sinf(ty * 0.5f);
          for (int i0 = 0; i0 < 16; ++i0) if (!(i0 & mask)) {
            int i1 = i0 | mask;
            float ar = sr[i0], ai = si[i0], br = sr[i1], bi = si[i1];
            sr[i0] = c * ar - s * br; si[i0] = c * ai - s * bi;
            sr[i1] = s * ar + c * br; si[i1] = s * ai + c * bi;
          }
        }
        { // RZ: diag(e, conj(e)), e = cos - i sin
          float c = cosf(tz * 0.5f), s = sinf(tz * 0.5f);
          for (int i0 = 0; i0 < 16; ++i0) if (!(i0 & mask)) {
            int i1 = i0 | mask;
            float ar = sr[i0], ai = si[i0], br = sr[i1], bi = si[i1];
            sr[i0] = c * ar + s * ai; si[i0] = c * ai - s * ar;
            sr[i1] = c * br - s * bi; si[i1] = c * bi + s * br;
          }
        }
      }
    }
    for (int i = 0; i < 16; ++i) {
      ws[WS_URE + g * 256 + i * 16 + col] = sr[i];
      ws[WS_UIM + g * 256 + i * 16 + col] = si[i];
    }
  }
}

// ---------------- main fused kernel ----------------
__global__ __launch_bounds__(256) void qlstm_main(
    const float* __restrict__ x, const float* __restrict__ h_prev,
    const float* __restrict__ c_prev, const float* __restrict__ ws,
    float* __restrict__ h_out, float* __restrict__ c_out)
{
  __shared__ float sW[16 * 644];              // stacked proj weights (padded)
  __shared__ float sComb[WAVES][16 * 68];     // combined chunk tile
  __shared__ float sAng[WAVES][16 * 20];      // 16x16 angle tile
  __shared__ float sV[WAVES][16 * 20];        // kron vectors
  __shared__ float sP[WAVES][4][16 * 20];     // probs per gate

  const int tid = threadIdx.x;
  const int wave = tid >> 5;
  const int lane = tid & 31;
  const int lr = lane & 15;   // M / N index within 16
  const int lh = lane >> 4;   // K half selector
  const int rowBase = (blockIdx.x * WAVES + wave) * 16;

  // cooperative load of stacked proj weights into LDS
  for (int v = tid; v < 16 * 160; v += 256) {
    int n = v / 160, k4 = (v - n * 160) * 4;
    *(float4*)(&sW[n * 644 + k4]) = *(const float4*)(ws + WS_W + n * COMB + k4);
  }
  __syncthreads();

  // ---- Phase 1: angle tile (16 rows x 16 = 4 gates x 4 qubits), K = 640 ----
  v8f accA = {0.f, 0.f, 0.f, 0.f, 0.f, 0.f, 0.f, 0.f};
  for (int ch = 0; ch < 10; ++ch) {
    const int k0 = ch * 64;
    {
      const int c4 = lr * 4;
      const int col = k0 + c4;
#pragma unroll
      for (int i = 0; i < 8; ++i) {
        int r = i * 2 + lh;
        float4 val;
        if (k0 < INSZ)
          val = *(const float4*)(x + (size_t)(rowBase + r) * INSZ + col);
        else
          val = *(const float4*)(h_prev + (size_t)(rowBase + r) * HID + (col - INSZ));
        *(float4*)(&sComb[wave][r * 68 + c4]) = val;
      }
    }
    __syncthreads();
#pragma unroll
    for (int kk = 0; kk < 16; ++kk) {
      v2f a = *(const v2f*)(&sComb[wave][lr * 68 + kk * 4 + 2 * lh]);
      v2f b = *(const v2f*)(&sW[lr * 644 + k0 + kk * 4 + 2 * lh]);
      accA = wmma4(a, b, accA);
    }
    __syncthreads();
  }
  {
    float pb = ws[WS_PB + lr];
#pragma unroll
    for (int j = 0; j < 8; ++j)
      sAng[wave][(j + 8 * lh) * 20 + lr] = accA[j] + pb;
  }
  __syncthreads();

  // ---- Phase 2: per gate, quantum layer via WMMA ----
  for (int g = 0; g < 4; ++g) {
    { // build kron vector V[row][16]
      float4 av = *(const float4*)(&sAng[wave][lr * 20 + g * 4]);
      float cq[4], sq[4];
      cq[0] = cosf(av.x * 0.5f); sq[0] = sinf(av.x * 0.5f);
      cq[1] = cosf(av.y * 0.5f); sq[1] = sinf(av.y * 0.5f);
      cq[2] = cosf(av.z * 0.5f); sq[2] = sinf(av.z * 0.5f);
      cq[3] = cosf(av.w * 0.5f); sq[3] = sinf(av.w * 0.5f);
      float vv[8];
#pragma unroll
      for (int t = 0; t < 8; ++t) {
        int c = 8 * lh + t;
        float p = ((c >> 3) & 1) ? sq[0] : cq[0];
        p *= ((c >> 2) & 1) ? sq[1] : cq[1];
        p *= ((c >> 1) & 1) ? sq[2] : cq[2];
        p *= (c & 1) ? sq[3] : cq[3];
        vv[t] = p;
      }
      *(float4*)(&sV[wave][lr * 20 + 8 * lh + 0]) = make_float4(vv[0], vv[1], vv[2], vv[3]);
      *(float4*)(&sV[wave][lr * 20 + 8 * lh + 4]) = make_float4(vv[4], vv[5], vv[6], vv[7]);
    }
    __syncthreads();

    v8f wre = {0.f, 0.f, 0.f, 0.f, 0.f, 0.f, 0.f, 0.f};
    v8f wim = {0.f, 0.f, 0.f, 0.f, 0.f, 0.f, 0.f, 0.f};
    const float* Ure = ws + WS_URE + g * 256;
    const float* Uim = ws + WS_UIM + g * 256;
#pragma unroll
    for (int kk = 0; kk < 4; ++kk) {
      v2f a  = *(const v2f*)(&sV[wave][lr * 20 + kk * 4 + 2 * lh]);
      v2f br = *(const v2f*)(Ure + lr * 16 + kk * 4 + 2 * lh);
      v2f bi = *(const v2f*)(Uim + lr * 16 + kk * 4 + 2 * lh);
      wre = wmma4(a, br, wre);
      wim = wmma4(a, bi, wim);
    }
#pragma unroll
    for (int j = 0; j < 8; ++j)
      sP[wave][g][(j + 8 * lh) * 20 + lr] = wre[j] * wre[j] + wim[j] * wim[j];
    __syncthreads();
  }

  // ---- Phase 3: fused output GEMMs (K=16) + LSTM combine ----
  const float* Mg = ws + WS_MGT;
  const float* lb = ws + WS_LB;
  for (int t = 0; t < 32; ++t) {
    v8f af = {0.f,0.f,0.f,0.f,0.f,0.f,0.f,0.f};
    v8f ai = {0.f,0.f,0.f,0.f,0.f,0.f,0.f,0.f};
    v8f ao = {0.f,0.f,0.f,0.f,0.f,0.f,0.f,0.f};
    v8f ag = {0.f,0.f,0.f,0.f,0.f,0.f,0.f,0.f};
    const int colB = t * 16 + lr;
#pragma unroll
    for (int kk = 0; kk < 4; ++kk) {
      const int aoff = lr * 20 + kk * 4 + 2 * lh;
      const int boff = colB * 16 + kk * 4 + 2 * lh;
      v2f a0 = *(const v2f*)(&sP[wave][0][aoff]);
      v2f a1 = *(const v2f*)(&sP[wave][1][aoff]);
      v2f a2 = *(const v2f*)(&sP[wave][2][aoff]);
      v2f a3 = *(const v2f*)(&sP[wave][3][aoff]);
      v2f b0 = *(const v2f*)(Mg + 0 * 8192 + boff);
      v2f b1 = *(const v2f*)(Mg + 1 * 8192 + boff);
      v2f b2 = *(const v2f*)(Mg + 2 * 8192 + boff);
      v2f b3 = *(const v2f*)(Mg + 3 * 8192 + boff);
      af = wmma4(a0, b0, af);
      ai = wmma4(a1, b1, ai);
      ao = wmma4(a2, b2, ao);
      ag = wmma4(a3, b3, ag);
    }
    const float bf = lb[0 * HID + colB];
    const float bi_ = lb[1 * HID + colB];
    const float bo = lb[2 * HID + colB];
    const float bg = lb[3 * HID + colB];
#pragma unroll
    for (int j = 0; j < 8; ++j) {
      size_t row = (size_t)(rowBase + j + 8 * lh);
      size_t off = row * HID + colB;
      float cp = c_prev[off];
      float fv = 1.f / (1.f + __expf(-(af[j] + bf)));
      float iv = 1.f / (1.f + __expf(-(ai[j] + bi_)));
      float ov = 1.f / (1.f + __expf(-(ao[j] + bo)));
      float gv = tanhf(ag[j] + bg);
      float cn = fv * cp + iv * gv;
      float hn = ov * tanhf(cn);
      h_out[off] = hn;
      c_out[off] = cn;
    }
  }
}

extern "C" void kernel_launch(void* const* d_in, const int* in_sizes, int n_in,
                              void* d_out, int out_size, void* d_ws, size_t ws_size,
                              hipStream_t stream) {
  (void)in_sizes; (void)n_in; (void)out_size; (void)ws_size;
  const float* x      = (const float*)d_in[0];
  const float* h_prev = (const float*)d_in[1];
  const float* c_prev = (const float*)d_in[2];
  float* ws = (float*)d_ws;

  qlstm_setup<<<1, 256, 0, stream>>>(
      (const float*)d_in[3],  (const float*)d_in[4],  (const float*)d_in[5],
      (const float*)d_in[6],  (const float*)d_in[7],
      (const float*)d_in[8],  (const float*)d_in[9],  (const float*)d_in[10],
      (const float*)d_in[11], (const float*)d_in[12],
      (const float*)d_in[13], (const float*)d_in[14], (const float*)d_in[15],
      (const float*)d_in[16], (const float*)d_in[17],
      (const float*)d_in[18], (const float*)d_in[19], (const float*)d_in[20],
      (const float*)d_in[21], (const float*)d_in[22],
      ws);

  float* h_out = (float*)d_out;
  float* c_out = h_out + (size_t)BATCH * HID;
  qlstm_main<<<BATCH / (WAVES * 16), 256, 0, stream>>>(x, h_prev, c_prev, ws, h_out, c_out);
}